// LocalGeometryEmbedding_83803401880230
// MI455X (gfx1250) — compile-verified
//
#include <hip/hip_runtime.h>
#include <hip/hip_bf16.h>

#define NPB   8192      // points per batch (hardcoded in reference)
#define KNN   10        // neighbors kept
#define FDIM  76        // feature dim
#define FPAD  80        // padded to multiple of 4 for WMMA K-loop
#define HDIM  64        // hidden dim

typedef __attribute__((ext_vector_type(2))) float v2f;
typedef __attribute__((ext_vector_type(8))) float v8f;

// ---------------------------------------------------------------------------
// Kernel 1: brute-force KNN. One thread per query point; candidates staged
// through LDS in tiles of 256. Sorted-ascending top-10 maintained in
// registers with a guarded unrolled insertion (common path = 1 compare).
// ---------------------------------------------------------------------------
__global__ __launch_bounds__(256)
void knn_kernel(const float* __restrict__ cloud, int* __restrict__ knn_out) {
    __shared__ float sx[256], sy[256], sz[256];
    const int tid = threadIdx.x;
    const int blocksPerBatch = NPB / 256;
    const int b = blockIdx.x / blocksPerBatch;
    const int i = (blockIdx.x % blocksPerBatch) * 256 + tid;
    const float* pc = cloud + (size_t)b * NPB * 3;

    const float qx = pc[i * 3 + 0];
    const float qy = pc[i * 3 + 1];
    const float qz = pc[i * 3 + 2];

    float bestd[KNN];
    int   besti[KNN];
#pragma unroll
    for (int k = 0; k < KNN; ++k) { bestd[k] = 3.4e38f; besti[k] = 0; }

    for (int base = 0; base < NPB; base += 256) {
        const int j0 = base + tid;
        const float tx = pc[j0 * 3 + 0];
        const float ty = pc[j0 * 3 + 1];
        const float tz = pc[j0 * 3 + 2];
        __syncthreads();
        sx[tid] = tx; sy[tid] = ty; sz[tid] = tz;
        __syncthreads();

        for (int t = 0; t < 256; ++t) {
            const int j = base + t;
            const float dx = sx[t] - qx;
            const float dy = sy[t] - qy;
            const float dz = sz[t] - qz;
            const float d2 = fmaf(dx, dx, fmaf(dy, dy, dz * dz));
            if (d2 < bestd[KNN - 1] && j != i) {
                // sorted insert (ascending); stable for ties since j increases
#pragma unroll
                for (int k = KNN - 1; k > 0; --k) {
                    if (d2 < bestd[k - 1]) {
                        bestd[k] = bestd[k - 1]; besti[k] = besti[k - 1];
                    } else if (d2 < bestd[k]) {
                        bestd[k] = d2; besti[k] = j;
                    }
                }
                if (d2 < bestd[0]) { bestd[0] = d2; besti[0] = j; }
            }
        }
    }

    int* out = knn_out + (size_t)(b * NPB + i) * KNN;
#pragma unroll
    for (int k = 0; k < KNN; ++k) out[k] = besti[k];
}

// ---------------------------------------------------------------------------
// Kernel 2: per-point 76-dim feature vector (padded to 80).
// center(3) | neighbors(30, k-major) | rel(30) | dists(10) | geom(3) | pad(4)
// Eigenvalues of the symmetric 3x3 covariance via the closed-form
// trigonometric method (matches eigvalsh ordering: l1 >= l2 >= l3).
// ---------------------------------------------------------------------------
__global__ __launch_bounds__(256)
void feat_kernel(const float* __restrict__ cloud, const int* __restrict__ knn_in,
                 float* __restrict__ feats) {
    const int gid = blockIdx.x * 256 + threadIdx.x;   // 0 .. B*NPB-1
    const int b = gid / NPB;
    const int i = gid - b * NPB;
    const float* pc = cloud + (size_t)b * NPB * 3;

    const float cx = pc[i * 3 + 0];
    const float cy = pc[i * 3 + 1];
    const float cz = pc[i * 3 + 2];

    float nx[KNN], ny[KNN], nz[KNN];
    const int* kid = knn_in + (size_t)gid * KNN;
#pragma unroll
    for (int k = 0; k < KNN; ++k) {
        const int j = kid[k];
        nx[k] = pc[j * 3 + 0];
        ny[k] = pc[j * 3 + 1];
        nz[k] = pc[j * 3 + 2];
    }

    // mean of neighbors
    float mx = 0.f, my = 0.f, mz = 0.f;
#pragma unroll
    for (int k = 0; k < KNN; ++k) { mx += nx[k]; my += ny[k]; mz += nz[k]; }
    mx *= 0.1f; my *= 0.1f; mz *= 0.1f;

    // covariance (unbiased, /(K-1))
    float a00 = 0.f, a01 = 0.f, a02 = 0.f, a11 = 0.f, a12 = 0.f, a22 = 0.f;
#pragma unroll
    for (int k = 0; k < KNN; ++k) {
        const float ux = nx[k] - mx, uy = ny[k] - my, uz = nz[k] - mz;
        a00 += ux * ux; a01 += ux * uy; a02 += ux * uz;
        a11 += uy * uy; a12 += uy * uz; a22 += uz * uz;
    }
    const float inv9 = 1.0f / 9.0f;
    a00 *= inv9; a01 *= inv9; a02 *= inv9;
    a11 *= inv9; a12 *= inv9; a22 *= inv9;

    // eigenvalues of symmetric 3x3
    const float q  = (a00 + a11 + a22) * (1.0f / 3.0f);
    const float p1 = a01 * a01 + a02 * a02 + a12 * a12;
    const float e0 = a00 - q, e1 = a11 - q, e2 = a22 - q;
    const float p2 = e0 * e0 + e1 * e1 + e2 * e2 + 2.0f * p1;
    float l1, l2, l3;
    if (p2 > 1e-30f) {
        const float p  = sqrtf(p2 * (1.0f / 6.0f));
        const float ip = 1.0f / p;
        const float b00 = e0 * ip, b11 = e1 * ip, b22 = e2 * ip;
        const float b01 = a01 * ip, b02 = a02 * ip, b12 = a12 * ip;
        float detB = b00 * (b11 * b22 - b12 * b12)
                   - b01 * (b01 * b22 - b12 * b02)
                   + b02 * (b01 * b12 - b11 * b02);
        float r = 0.5f * detB;
        r = fminf(1.0f, fmaxf(-1.0f, r));
        const float phi = acosf(r) * (1.0f / 3.0f);
        l1 = q + 2.0f * p * cosf(phi);
        l3 = q + 2.0f * p * cosf(phi + 2.0943951023931953f); // +2*pi/3
        l2 = 3.0f * q - l1 - l3;
    } else {
        l1 = l2 = l3 = q;
    }
    const float il1 = 1.0f / l1;
    const float lin  = (l1 - l2) * il1;
    const float plan = (l2 - l3) * il1;
    const float scat = l3 * il1;

    float* f = feats + (size_t)gid * FPAD;
    f[0] = cx; f[1] = cy; f[2] = cz;
#pragma unroll
    for (int k = 0; k < KNN; ++k) {
        const float rx = nx[k] - cx, ry = ny[k] - cy, rz = nz[k] - cz;
        f[3 + 3 * k + 0]  = nx[k];
        f[3 + 3 * k + 1]  = ny[k];
        f[3 + 3 * k + 2]  = nz[k];
        f[33 + 3 * k + 0] = rx;
        f[33 + 3 * k + 1] = ry;
        f[33 + 3 * k + 2] = rz;
        f[63 + k] = sqrtf(fmaf(rx, rx, fmaf(ry, ry, rz * rz)));
    }
    f[73] = lin; f[74] = plan; f[75] = scat;
    f[76] = 0.f; f[77] = 0.f; f[78] = 0.f; f[79] = 0.f;
}

// ---------------------------------------------------------------------------
// Kernel 3: fused MLP via V_WMMA_F32_16X16X4_F32 (fp32, exact-precision path).
// Block = 128 threads = 4 waves; each wave owns a 32-row M tile (2 A frags
// per B frag -> every ds_load_b64 of B feeds two WMMAs).
// B operands packed as float2 pairs of consecutive K rows so each fragment
// is one aligned ds_load_b64 directly into the operand VGPR pair.
// GEMM1: [32x80]x[80x64] = 20 k-steps x 4 N-tiles x 2 M-tiles = 160 WMMA.
// GEMM2: [32x64]x[64x16(pad of 3)] = 16 k-steps x 2 M-tiles    =  32 WMMA.
// ---------------------------------------------------------------------------
__global__ __launch_bounds__(128)
void mlp_kernel(const float* __restrict__ feats,
                const float* __restrict__ W1, const float* __restrict__ b1,
                const float* __restrict__ W2, const float* __restrict__ b2,
                float* __restrict__ out) {
    __shared__ v2f   sW1p[(FPAD / 2) * HDIM];  // [40][64] pairs of K rows
    __shared__ v2f   sW2p[(HDIM / 2) * 16];    // [32][16] pairs, cols 3..15 = 0
    __shared__ float sB1[HDIM];
    __shared__ float sB2[16];
    __shared__ float sH[4][32 * HDIM];         // per-wave ReLU'd hidden tile

    const int tid = threadIdx.x;
    for (int idx = tid; idx < (FPAD / 2) * HDIM; idx += 128) {
        const int p = idx / HDIM, n = idx - p * HDIM;
        v2f w;
        w.x = (2 * p + 0 < FDIM) ? W1[(2 * p + 0) * HDIM + n] : 0.0f;
        w.y = (2 * p + 1 < FDIM) ? W1[(2 * p + 1) * HDIM + n] : 0.0f;
        sW1p[idx] = w;
    }
    for (int idx = tid; idx < (HDIM / 2) * 16; idx += 128) {
        const int p = idx >> 4, o = idx & 15;
        v2f w;
        w.x = (o < 3) ? W2[(2 * p + 0) * 3 + o] : 0.0f;
        w.y = (o < 3) ? W2[(2 * p + 1) * 3 + o] : 0.0f;
        sW2p[idx] = w;
    }
    if (tid < HDIM) sB1[tid] = b1[tid];
    if (tid < 16)   sB2[tid] = (tid < 3) ? b2[tid] : 0.0f;
    __syncthreads();

    const int wv   = tid >> 5;
    const int lane = tid & 31;
    const int m    = lane & 15;       // column (B/C/D) or row (A)
    const int hi   = lane >> 4;       // half-select per 16x4 A/B layout
    const int row0 = blockIdx.x * 128 + wv * 32;
    const float* frow0 = feats + (size_t)(row0 + m) * FPAD;
    const float* frow1 = feats + (size_t)(row0 + 16 + m) * FPAD;

    // ---- GEMM1: feats x W1 -> h[32x64] ----
    v8f acc[2][4] = {};
    for (int kb = 0; kb < FPAD; kb += 4) {
        const v2f a0 = *(const v2f*)(frow0 + kb + 2 * hi);
        const v2f a1 = *(const v2f*)(frow1 + kb + 2 * hi);
        const int prow = (kb >> 1) + hi;
#pragma unroll
        for (int t = 0; t < 4; ++t) {
            const v2f bf = sW1p[prow * HDIM + t * 16 + m];
            acc[0][t] = __builtin_amdgcn_wmma_f32_16x16x4_f32(
                false, a0, false, bf, (short)0, acc[0][t], false, false);
            acc[1][t] = __builtin_amdgcn_wmma_f32_16x16x4_f32(
                false, a1, false, bf, (short)0, acc[1][t], false, false);
        }
    }

    // bias + ReLU; re-fragment through per-wave LDS region (same-wave DS ops
    // are ordered, so no block barrier needed)
    float* hb = &sH[wv][0];
#pragma unroll
    for (int g = 0; g < 2; ++g) {
#pragma unroll
        for (int t = 0; t < 4; ++t) {
            const int n = t * 16 + m;
            const float bias = sB1[n];
#pragma unroll
            for (int r = 0; r < 8; ++r) {
                const int row = g * 16 + r + 8 * hi;
                hb[row * HDIM + n] = fmaxf(acc[g][t][r] + bias, 0.0f);
            }
        }
    }

    // ---- GEMM2: h x W2(pad16) -> out[32x3] ----
    const v2f* hb2 = (const v2f*)hb;   // pairs of consecutive hidden cols
    v8f acc2[2] = {};
    for (int kb = 0; kb < HDIM; kb += 4) {
        const v2f bf2 = sW2p[((kb >> 1) + hi) * 16 + m];
        const v2f a20 = hb2[(m)      * (HDIM / 2) + (kb >> 1) + hi];
        const v2f a21 = hb2[(16 + m) * (HDIM / 2) + (kb >> 1) + hi];
        acc2[0] = __builtin_amdgcn_wmma_f32_16x16x4_f32(
            false, a20, false, bf2, (short)0, acc2[0], false, false);
        acc2[1] = __builtin_amdgcn_wmma_f32_16x16x4_f32(
            false, a21, false, bf2, (short)0, acc2[1], false, false);
    }

    if (m < 3) {
        const float bias = sB2[m];
#pragma unroll
        for (int g = 0; g < 2; ++g) {
#pragma unroll
            for (int r = 0; r < 8; ++r) {
                const int row = row0 + g * 16 + r + 8 * hi;
                out[row * 3 + m] = fmaxf(acc2[g][r] + bias, 0.0f);
            }
        }
    }
}

// ---------------------------------------------------------------------------
extern "C" void kernel_launch(void* const* d_in, const int* in_sizes, int n_in,
                              void* d_out, int out_size, void* d_ws, size_t ws_size,
                              hipStream_t stream) {
    const float* cloud = (const float*)d_in[0];
    const float* W1    = (const float*)d_in[1];
    const float* b1    = (const float*)d_in[2];
    const float* W2    = (const float*)d_in[3];
    const float* b2    = (const float*)d_in[4];
    float* out = (float*)d_out;

    const int totalPts = in_sizes[0] / 3;          // B * NPB (= 32768)

    int*   knn   = (int*)d_ws;
    float* feats = (float*)((char*)d_ws + (size_t)totalPts * KNN * sizeof(int));

    knn_kernel <<<totalPts / 256, 256, 0, stream>>>(cloud, knn);
    feat_kernel<<<totalPts / 256, 256, 0, stream>>>(cloud, knn, feats);
    mlp_kernel <<<totalPts / 128, 128, 0, stream>>>(feats, W1, b1, W2, b2, out);
}